// Attntion_33097017983020
// MI455X (gfx1250) — compile-verified
//
#include <hip/hip_runtime.h>
#include <math.h>
#include <stdint.h>

// Problem sizes are fixed by the reference: B=4, N1=N2=1024, D=256, H=4, Dh=64.
#define B_  4
#define N1_ 1024
#define N2_ 1024
#define D_  256
#define H_  4
#define DH_ 64

typedef __attribute__((ext_vector_type(2))) float v2f;
typedef __attribute__((ext_vector_type(8))) float v8f;

// ---------------------------------------------------------------- helpers ---
__device__ __forceinline__ float gelu_tanh(float x) {
  const float c0 = 0.7978845608028654f;  // sqrt(2/pi)
  const float c1 = 0.044715f;
  return 0.5f * x * (1.0f + tanhf(c0 * (x + c1 * x * x * x)));
}

// wave32 shuffle reductions
__device__ __forceinline__ float waveSum(float v) {
#pragma unroll
  for (int o = 16; o > 0; o >>= 1) v += __shfl_down(v, o, 32);
  return v;
}
__device__ __forceinline__ float waveMax(float v) {
#pragma unroll
  for (int o = 16; o > 0; o >>= 1) v = fmaxf(v, __shfl_down(v, o, 32));
  return v;
}
// blockDim.x == 128 (4 wave32s) in every kernel that calls these.
__device__ __forceinline__ float blockSum(float v, float* s) {
  v = waveSum(v);
  __syncthreads();
  if ((threadIdx.x & 31) == 0) s[threadIdx.x >> 5] = v;
  __syncthreads();
  return s[0] + s[1] + s[2] + s[3];
}
__device__ __forceinline__ float blockMax(float v, float* s) {
  v = waveMax(v);
  __syncthreads();
  if ((threadIdx.x & 31) == 0) s[threadIdx.x >> 5] = v;
  __syncthreads();
  return fmaxf(fmaxf(s[0], s[1]), fmaxf(s[2], s[3]));
}

// ------------------------------------------------------------------- k0 -----
// bt_pre = prev @ W_pre + b_pre ; bt_cur = curr @ W_cur + b_cur.
__global__ void k0_bt(const float* __restrict__ prev, const float* __restrict__ curr,
                      const float* __restrict__ W_pre, const float* __restrict__ b_pre,
                      const float* __restrict__ W_cur, const float* __restrict__ b_cur,
                      float* __restrict__ bt_pre, float* __restrict__ bt_cur) {
  int gw   = (blockIdx.x * blockDim.x + threadIdx.x) >> 5;  // global wave id
  int lane = threadIdx.x & 31;
  const int ROWS = B_ * N1_;
  const float *x, *W, *bias;
  float* outp;
  if (gw < ROWS) {
    x = prev + (size_t)gw * D_; W = W_pre; bias = b_pre; outp = bt_pre + (size_t)gw * 6;
  } else {
    int r = gw - ROWS;
    x = curr + (size_t)r * D_;  W = W_cur; bias = b_cur; outp = bt_cur + (size_t)r * 6;
  }
  float acc[6] = {0, 0, 0, 0, 0, 0};
  for (int i = lane; i < D_; i += 32) {
    float xi = x[i];
#pragma unroll
    for (int j = 0; j < 6; ++j) acc[j] += xi * W[i * 6 + j];
  }
#pragma unroll
  for (int j = 0; j < 6; ++j) {
    float s = waveSum(acc[j]);
    if (lane == 0) outp[j] = s + bias[j];
  }
}

// ------------------------------------------------------------------- k1 -----
// prevT[b][d][n1] = prev[b][n1][d]  (so k4's B-fragment is k-contiguous).
__global__ void k1_transpose(const float* __restrict__ prev, float* __restrict__ prevT) {
  __shared__ float tile[32][33];
  int b = blockIdx.z;
  int n1_0 = blockIdx.x * 32;
  int d_0  = blockIdx.y * 32;
  int c = threadIdx.x & 31;
  int r0 = threadIdx.x >> 5;   // 8 rows per pass, 4 passes (256 threads)
#pragma unroll
  for (int i = 0; i < 4; ++i) {
    int r = r0 + i * 8;
    tile[r][c] = prev[((size_t)b * N1_ + n1_0 + r) * D_ + d_0 + c];
  }
  __syncthreads();
#pragma unroll
  for (int i = 0; i < 4; ++i) {
    int r = r0 + i * 8;
    prevT[((size_t)b * D_ + d_0 + r) * N1_ + n1_0 + c] = tile[c][r];
  }
}

// ------------------------------------------------------------------- k2 -----
// Per (b,n2) column: colsum/colmax of masked c, e1inv = 1/(sum edges + 1).
__global__ void k2_colstats(const float* __restrict__ bt_pre, const float* __restrict__ bt_cur,
                            const int* __restrict__ edges,
                            const float* __restrict__ W_fuse, const float* __restrict__ b_fuse,
                            float* __restrict__ colsum, float* __restrict__ colmax,
                            float* __restrict__ e1inv) {
  __shared__ float sred[4];
  int n2 = blockIdx.x % N2_;
  int b  = blockIdx.x / N2_;
  int t  = threadIdx.x;

  float ccur[4];
  {
    const float* btc = bt_cur + ((size_t)b * N2_ + n2) * 6;
    float g[6];
#pragma unroll
    for (int i = 0; i < 6; ++i) g[i] = gelu_tanh(btc[i]);
#pragma unroll
    for (int j = 0; j < 4; ++j) {
      float s = b_fuse[j];
#pragma unroll
      for (int i = 0; i < 6; ++i) s += g[i] * W_fuse[(6 + i) * 4 + j];
      ccur[j] = s;
    }
  }

  float sum4[4] = {0, 0, 0, 0};
  float max4[4] = {-__builtin_inff(), -__builtin_inff(), -__builtin_inff(), -__builtin_inff()};
  float cnt = 0.0f;
  for (int n1 = t; n1 < N1_; n1 += 128) {
    const float* btp = bt_pre + ((size_t)b * N1_ + n1) * 6;
    int e = edges[((size_t)b * N1_ + n1) * N2_ + n2];
    float gp[6];
#pragma unroll
    for (int i = 0; i < 6; ++i) gp[i] = gelu_tanh(btp[i]);
#pragma unroll
    for (int j = 0; j < 4; ++j) {
      float c = ccur[j];
#pragma unroll
      for (int i = 0; i < 6; ++i) c += gp[i] * W_fuse[i * 4 + j];
      float mc = e ? c : 0.0f;
      sum4[j] += mc;
      max4[j] = fmaxf(max4[j], mc);
    }
    cnt += e ? 1.0f : 0.0f;
  }
#pragma unroll
  for (int j = 0; j < 4; ++j) {
    float s = blockSum(sum4[j], sred);
    float m = blockMax(max4[j], sred);
    if (t == 0) {
      colsum[((size_t)b * N2_ + n2) * 4 + j] = s;
      colmax[((size_t)b * N2_ + n2) * 4 + j] = m;
    }
  }
  float ct = blockSum(cnt, sred);
  if (t == 0) e1inv[(size_t)b * N2_ + n2] = 1.0f / (ct + 1.0f);
}

// ------------------------------------------------------------------- k3 -----
// Per (b,n2) column: recompute c, build fused & att in registers, block-reduce
// softmax max, store UNNORMALIZED exp(att-m) head-major into
// ebuf[B][H][N2][N1] (k-contiguous for k4), store 1/l into rl[B][H][N2],
// and emit gbias = gelu(g) @ W_fcg + b_fcg.
__global__ void k3_fused(const float* __restrict__ bt_pre, const float* __restrict__ bt_cur,
                         const int* __restrict__ edges,
                         const float* __restrict__ W_fuse, const float* __restrict__ b_fuse,
                         const float* __restrict__ W_ffn,  const float* __restrict__ b_ffn,
                         const float* __restrict__ W_fcc,  const float* __restrict__ b_fcc,
                         const float* __restrict__ W_fcg,  const float* __restrict__ b_fcg,
                         const float* __restrict__ colsum, const float* __restrict__ colmax,
                         const float* __restrict__ e1inv_,
                         float* __restrict__ ebuf, float* __restrict__ rl,
                         float* __restrict__ gbias) {
  __shared__ float sred[4];
  int n2 = blockIdx.x % N2_;
  int b  = blockIdx.x / N2_;
  int t  = threadIdx.x;
  float einv = e1inv_[(size_t)b * N2_ + n2];

  float ccur[4];
  {
    const float* btc = bt_cur + ((size_t)b * N2_ + n2) * 6;
    float g[6];
#pragma unroll
    for (int i = 0; i < 6; ++i) g[i] = gelu_tanh(btc[i]);
#pragma unroll
    for (int j = 0; j < 4; ++j) {
      float s = b_fuse[j];
#pragma unroll
      for (int i = 0; i < 6; ++i) s += g[i] * W_fuse[(6 + i) * 4 + j];
      ccur[j] = s;
    }
  }
  float base8[8];
  {
    const float* cs = colsum + ((size_t)b * N2_ + n2) * 4;
    const float* cm = colmax + ((size_t)b * N2_ + n2) * 4;
    float gd[4], ge[4];
#pragma unroll
    for (int j = 0; j < 4; ++j) {
      gd[j] = gelu_tanh(cs[j] * einv);
      ge[j] = gelu_tanh(cm[j]);
    }
#pragma unroll
    for (int k = 0; k < 8; ++k) {
      float s = b_ffn[k];
#pragma unroll
      for (int i = 0; i < 4; ++i)
        s += gd[i] * W_ffn[(4 + i) * 8 + k] + ge[i] * W_ffn[(8 + i) * 8 + k];
      base8[k] = s;
    }
  }

  float attr[8][4];
  float mmax[4] = {-__builtin_inff(), -__builtin_inff(), -__builtin_inff(), -__builtin_inff()};
  float gsum[8] = {0, 0, 0, 0, 0, 0, 0, 0};
#pragma unroll
  for (int it = 0; it < 8; ++it) {
    int n1 = t + it * 128;
    const float* btp = bt_pre + ((size_t)b * N1_ + n1) * 6;
    int e = edges[((size_t)b * N1_ + n1) * N2_ + n2];
    float gp[6];
#pragma unroll
    for (int i = 0; i < 6; ++i) gp[i] = gelu_tanh(btp[i]);
    float gc[4];
#pragma unroll
    for (int j = 0; j < 4; ++j) {
      float c = ccur[j];
#pragma unroll
      for (int i = 0; i < 6; ++i) c += gp[i] * W_fuse[i * 4 + j];
      c = e ? c : 0.0f;
      gc[j] = gelu_tanh(c);
    }
    float gf[8];
#pragma unroll
    for (int k = 0; k < 8; ++k) {
      float s = base8[k];
#pragma unroll
      for (int j = 0; j < 4; ++j) s += gc[j] * W_ffn[j * 8 + k];
      if (e) gsum[k] += s;
      gf[k] = gelu_tanh(s);
    }
#pragma unroll
    for (int j = 0; j < 4; ++j) {
      float a = b_fcc[j];
#pragma unroll
      for (int k = 0; k < 8; ++k) a += gf[k] * W_fcc[k * 4 + j];
      a = e ? a : -__builtin_inff();      // masked logit -> exp gives 0
      attr[it][j] = a;
      mmax[j] = fmaxf(mmax[j], a);
    }
  }

  float m[4];
#pragma unroll
  for (int j = 0; j < 4; ++j) m[j] = fmaxf(blockMax(mmax[j], sred), -1e30f);

  float lsum[4] = {0, 0, 0, 0};
#pragma unroll
  for (int it = 0; it < 8; ++it) {
    int n1 = t + it * 128;
#pragma unroll
    for (int j = 0; j < 4; ++j) {
      float ev = __expf(attr[it][j] - m[j]);
      ebuf[(((size_t)b * H_ + j) * N2_ + n2) * N1_ + n1] = ev;  // head-major
      lsum[j] += ev;
    }
  }
#pragma unroll
  for (int j = 0; j < 4; ++j) {
    float l = blockSum(lsum[j], sred);
    if (t == 0) rl[((size_t)b * H_ + j) * N2_ + n2] = (l > 0.0f) ? (1.0f / l) : 0.0f;
  }

  float g8[8];
#pragma unroll
  for (int k = 0; k < 8; ++k) {
    float s = blockSum(gsum[k], sred);
    g8[k] = gelu_tanh(s * einv);
  }
  float* gb = gbias + ((size_t)b * N2_ + n2) * D_;
  for (int d = t; d < D_; d += 128) {
    float s = b_fcg[d];
#pragma unroll
    for (int k = 0; k < 8; ++k) s += g8[k] * W_fcg[k * D_ + d];
    gb[d] = s;
  }
}

// ------------------------------------------------------------------- k4 -----
// res = einsum('bmnh,bnhd', p, prev_head) via V_WMMA_F32_16X16X4_F32.
// Block = (b,h,n2-tile); its 16x1024 A tile (softmax numerators) is SHARED by
// all 4 waves, so it is staged once into LDS with GLOBAL_LOAD_ASYNC_TO_LDS_B128
// (ASYNCcnt), double-buffered in 256-wide K slabs and consumed via ds_load_b64.
// B fragments come from the k-contiguous prevT as single b64 loads.
// Fragment layouts per ISA §7.12.2:
//   A 16x4: M = lane&15; lanes<16 hold K={k,k+1}, lanes>=16 hold K={k+2,k+3}.
//   B 4x16: N = lane&15; lanes<16 hold rows K={k,k+1}, lanes>=16 K={k+2,k+3}.
//   C/D:    reg j -> M=j (lanes<16) / M=8+j (lanes>=16), N = lane&15.
#define KSLAB 256
__global__ void k4_aggregate(const float* __restrict__ ebufH, const float* __restrict__ rlH,
                             const float* __restrict__ prevT, const float* __restrict__ gbias,
                             float* __restrict__ out) {
  __shared__ float smem[2][16][KSLAB];   // 2 x 16KB A slabs
  int blk  = blockIdx.x;
  int n2t  = blk & (N2_ / 16 - 1);
  int h    = (blk >> 6) & (H_ - 1);
  int b    = blk >> 8;
  int wave = threadIdx.x >> 5;
  int lane = threadIdx.x & 31;
  int half = lane >> 4;
  int l15  = lane & 15;
  int n2_0 = n2t * 16;
  int d0   = h * DH_ + wave * 16;

  const float* eplane = ebufH + ((size_t)b * H_ + h) * (size_t)N2_ * N1_;  // + n2*N1 + n1
  const float* pT     = prevT + ((size_t)b * D_ + (d0 + l15)) * N1_;       // k-contiguous
  float scaleA = rlH[((size_t)b * H_ + h) * N2_ + (n2_0 + l15)];

  // Issue one 16x256 A slab (16KB) as 8 per-thread async b128 copies.
  auto issue_slab = [&](int s, int buf) {
    int t = threadIdx.x;
#pragma unroll
    for (int i = 0; i < 8; ++i) {
      int q   = t + i * 128;        // 0..1023 16B-chunks
      int row = q >> 6;             // 0..15  (64 chunks per 1KB row)
      int c16 = q & 63;
      const float* ga = eplane + (size_t)(n2_0 + row) * N1_ + s * KSLAB + c16 * 4;
      unsigned lofs = (unsigned)(uintptr_t)&smem[buf][row][c16 * 4];  // LDS byte offset
      asm volatile("global_load_async_to_lds_b128 %0, %1, off"
                   :: "v"(lofs), "v"((uint64_t)(uintptr_t)ga)
                   : "memory");
    }
  };

  v8f acc = {};
  issue_slab(0, 0);
  for (int s = 0; s < N1_ / KSLAB; ++s) {
    if (s < N1_ / KSLAB - 1) {
      issue_slab(s + 1, (s + 1) & 1);                    // prefetch next slab
      asm volatile("s_wait_asynccnt 0x8" ::: "memory");  // slab s landed (in-order)
    } else {
      asm volatile("s_wait_asynccnt 0x0" ::: "memory");
    }
    __syncthreads();

    const float* As = &smem[s & 1][l15][0];
    const float* Bs = pT + s * KSLAB;
    __builtin_prefetch(Bs + KSLAB, 0, 1);                // global_prefetch_b8 next B slab
#pragma unroll 8
    for (int kk = 0; kk < KSLAB; kk += 4) {
      int ka = kk + half * 2;
      v2f A = *(const v2f*)(As + ka);                    // ds_load_b64
      A.x *= scaleA;                                     // p = exp * (1/l)
      A.y *= scaleA;
      v2f Bf = *(const v2f*)(Bs + ka);                   // global_load_b64
      acc = __builtin_amdgcn_wmma_f32_16x16x4_f32(false, A, false, Bf,
                                                  (short)0, acc, false, false);
    }
    __syncthreads();   // all waves done reading buf (s&1) before it is refilled
  }

  size_t obase = (size_t)b * N2_ * D_;
  int n2r_base = n2_0 + half * 8;
#pragma unroll
  for (int j = 0; j < 8; ++j) {
    size_t idx = obase + (size_t)(n2r_base + j) * D_ + (d0 + l15);
    out[idx] = acc[j] + gbias[idx];
  }
}

// ------------------------------------------------------------- launcher -----
extern "C" void kernel_launch(void* const* d_in, const int* in_sizes, int n_in,
                              void* d_out, int out_size, void* d_ws, size_t ws_size,
                              hipStream_t stream) {
  const float* prev   = (const float*)d_in[0];
  const float* curr   = (const float*)d_in[1];
  const int*   edges  = (const int*)  d_in[2];
  const float* W_pre  = (const float*)d_in[3];
  const float* b_pre  = (const float*)d_in[4];
  const float* W_cur  = (const float*)d_in[5];
  const float* b_cur  = (const float*)d_in[6];
  const float* W_fuse = (const float*)d_in[7];
  const float* b_fuse = (const float*)d_in[8];
  const float* W_ffn  = (const float*)d_in[9];
  const float* b_ffn  = (const float*)d_in[10];
  const float* W_fcc  = (const float*)d_in[11];
  const float* b_fcc  = (const float*)d_in[12];
  const float* W_fcg  = (const float*)d_in[13];
  const float* b_fcg  = (const float*)d_in[14];
  float* out = (float*)d_out;

  // workspace layout (floats): ~76 MB total
  float* w = (float*)d_ws;
  size_t off = 0;
  float* bt_pre = w + off; off += (size_t)B_ * N1_ * 6;
  float* bt_cur = w + off; off += (size_t)B_ * N2_ * 6;
  float* e1inv  = w + off; off += (size_t)B_ * N2_;
  float* colsum = w + off; off += (size_t)B_ * N2_ * 4;
  float* colmax = w + off; off += (size_t)B_ * N2_ * 4;
  float* rl     = w + off; off += (size_t)B_ * H_ * N2_;
  float* gbias  = w + off; off += (size_t)B_ * N2_ * D_;
  float* prevT  = w + off; off += (size_t)B_ * D_ * N1_;
  float* ebuf   = w + off; off += (size_t)B_ * H_ * N2_ * N1_;  // 64MB, fits in L2

  k0_bt<<<2048, 128, 0, stream>>>(prev, curr, W_pre, b_pre, W_cur, b_cur, bt_pre, bt_cur);
  k1_transpose<<<dim3(N1_ / 32, D_ / 32, B_), 256, 0, stream>>>(prev, prevT);
  k2_colstats<<<B_ * N2_, 128, 0, stream>>>(bt_pre, bt_cur, edges, W_fuse, b_fuse,
                                            colsum, colmax, e1inv);
  k3_fused<<<B_ * N2_, 128, 0, stream>>>(bt_pre, bt_cur, edges, W_fuse, b_fuse,
                                         W_ffn, b_ffn, W_fcc, b_fcc, W_fcg, b_fcg,
                                         colsum, colmax, e1inv, ebuf, rl, gbias);
  k4_aggregate<<<B_ * H_ * (N2_ / 16), 128, 0, stream>>>(ebuf, rl, prevT, gbias, out);
}